// CorruptionPipeline_7438883357302
// MI455X (gfx1250) — compile-verified
//
#include <hip/hip_runtime.h>
#include <stdint.h>

// CorruptionPipeline for MI455X (gfx1250).
// Bandwidth-bound (512 MiB HBM traffic -> ~22us floor @ 23.3 TB/s).
// One workgroup per row. Symmetric Tensor-Data-Mover staging:
//   TDM load  : Global row -> LDS (TENSORcnt)
//   compute   : Philox mask/noise, in-LDS bitonic argsort perm, swap-gather
//   TDM store : LDS staged row -> Global (reuses dead sort-key buffer)
// Row-local work never touches the vector memory pipe; HBM traffic is pure DMA.

#define FDIM   2048
#define BLOCK  256
#define EPT    (FDIM / BLOCK)   // 8 elements per thread

typedef __attribute__((ext_vector_type(4))) unsigned int u32x4;
typedef __attribute__((ext_vector_type(8))) int          i32x8;
typedef __attribute__((ext_vector_type(4))) int          i32x4;

#if defined(__has_builtin)
#if __has_builtin(__builtin_amdgcn_tensor_load_to_lds) && \
    __has_builtin(__builtin_amdgcn_tensor_store_from_lds)
#define HAVE_TDM 1
#endif
#endif

// ---------------- Philox 4x32-10 counter RNG (deterministic) ----------------
__device__ __forceinline__ void philox4(uint32_t c0, uint32_t c1, uint32_t c2,
                                        uint32_t c3, uint32_t k0, uint32_t k1,
                                        uint32_t out[4]) {
  const uint32_t M0 = 0xD2511F53u, M1 = 0xCD9E8D57u;
  const uint32_t W0 = 0x9E3779B9u, W1 = 0xBB67AE85u;
#pragma unroll
  for (int r = 0; r < 10; ++r) {
    uint32_t h0 = __umulhi(M0, c0), l0 = M0 * c0;
    uint32_t h1 = __umulhi(M1, c2), l1 = M1 * c2;
    uint32_t n0 = h1 ^ c1 ^ k0;
    uint32_t n2 = h0 ^ c3 ^ k1;
    c0 = n0; c1 = l1; c2 = n2; c3 = l0;
    k0 += W0; k1 += W1;
  }
  out[0] = c0; out[1] = c1; out[2] = c2; out[3] = c3;
}

__device__ __forceinline__ float u01(uint32_t v) {
  // uniform in (0,1] -- safe for logf in Box-Muller
  return (float)((v >> 8) + 1u) * (1.0f / 16777216.0f);
}

struct RowSmem {
  float    val[FDIM];   // LDS offset 0    : TDM load target / corrupted values
  float    key[FDIM];   // LDS offset 8192 : sort keys, then output staging
  uint32_t idx[FDIM];   // LDS offset 16384: permutation payload
};

#ifdef HAVE_TDM
// D# for a 1-row (2048 x f32) tile: count=1, data_size=4B, tensor_dim0 =
// tile_dim0 = 2048, tensor_dim1 = tile_dim1 = 1, stride0 = 2048, type=2.
__device__ __forceinline__ void make_row_desc(unsigned lds_off,
                                              unsigned long long ga,
                                              u32x4& g0, i32x8& g1) {
  g0 = { 1u,
         lds_off,
         (unsigned)(ga & 0xffffffffu),
         (unsigned)((ga >> 32) & 0x01ffffffu) | (2u << 30) };
  g1 = { (int)(2u << 16),                        // data_size = 4 bytes
         (int)(((unsigned)FDIM & 0xffffu) << 16),// tensor_dim0.lo16
         (int)(1u << 16),                        // tensor_dim0.hi16=0 | tensor_dim1=1
         (int)((unsigned)FDIM << 16),            // tensor_dim1.hi16=0 | tile_dim0
         1,                                      // tile_dim1=1, tile_dim2=0
         (int)FDIM,                              // tensor_dim0_stride.lo32
         0, 0 };
}
#endif

__global__ __launch_bounds__(BLOCK)
void corruption_pipeline_7438883357302(const float* __restrict__ x,
                                       float* __restrict__ out) {
  __shared__ RowSmem sm;

  const uint32_t row = blockIdx.x;
  const uint32_t tid = threadIdx.x;
  const size_t   rowBase = (size_t)row * FDIM;

  // ---------------- Stage 0: row Global -> LDS via Tensor Data Mover -------
#ifdef HAVE_TDM
  if (tid < 32) {  // wave 0 only issues the DMA (TDM ignores EXEC)
    u32x4 g0; i32x8 g1; i32x4 gz = { 0, 0, 0, 0 };
    make_row_desc(0u, (unsigned long long)(const void*)(x + rowBase), g0, g1);
#if __has_include(<hip/amd_detail/amd_gfx1250_TDM.h>)
    i32x8 gz8 = { 0, 0, 0, 0, 0, 0, 0, 0 };
    __builtin_amdgcn_tensor_load_to_lds(g0, g1, gz, gz, gz8, 0); // clang-23 6-arg
#else
    __builtin_amdgcn_tensor_load_to_lds(g0, g1, gz, gz, 0);      // ROCm7.2 5-arg
#endif
    __builtin_amdgcn_s_wait_tensorcnt((short)0);
  }
#else
  {  // fallback: cooperative vector load
    const float4* src = (const float4*)(x + rowBase);
    for (uint32_t i = tid; i < FDIM / 4; i += BLOCK) {
      float4 v = src[i];
      sm.val[4 * i + 0] = v.x; sm.val[4 * i + 1] = v.y;
      sm.val[4 * i + 2] = v.z; sm.val[4 * i + 3] = v.w;
    }
  }
#endif
  __syncthreads();

  // ---------------- Stage 1: mask -> 0, add Gaussian noise, make perm keys --
  const uint32_t base = tid * EPT;
#pragma unroll
  for (int q = 0; q < EPT; q += 4) {
    const uint32_t col = base + (uint32_t)q;
    uint32_t rm[4], rn[4], rk[4];
    philox4(row, col, 0u, 0u, 42u, 0x6d61736bu, rm);  // "mask"
    philox4(row, col, 1u, 0u, 42u, 0x6e6f6973u, rn);  // "nois"
    philox4(row, col, 3u, 0u, 42u, 0x7065726du, rk);  // "perm"

    // Box-Muller: two pairs of uniforms -> four normals
    float r1 = sqrtf(-2.0f * __logf(u01(rn[0])));
    float r2 = sqrtf(-2.0f * __logf(u01(rn[2])));
    float s1, c1, s2, c2;
    __sincosf(6.28318530718f * u01(rn[1]), &s1, &c1);
    __sincosf(6.28318530718f * u01(rn[3]), &s2, &c2);
    float nrm[4] = { r1 * c1, r1 * s1, r2 * c2, r2 * s2 };

#pragma unroll
    for (int t = 0; t < 4; ++t) {
      const uint32_t j = col + (uint32_t)t;
      float v = sm.val[j];
      v = (u01(rm[t]) > 0.15f) ? v : 0.0f;  // RandomMasking (keep if > rate)
      v = fmaf(0.1f, nrm[t], v);            // GaussianNoise
      sm.val[j] = v;
      sm.key[j] = u01(rk[t]);               // argsort keys
      sm.idx[j] = j;
    }
  }
  __syncthreads();

  // ---------------- Stage 2: per-row permutation = argsort(uniform) --------
  // In-LDS bitonic sort of 2048 (key,idx) pairs, index tie-break (stable).
  for (uint32_t k = 2; k <= FDIM; k <<= 1) {
    for (uint32_t j = k >> 1; j > 0; j >>= 1) {
#pragma unroll 1
      for (uint32_t a = tid; a < FDIM; a += BLOCK) {
        const uint32_t b = a ^ j;
        if (b > a) {
          const bool up = ((a & k) == 0);
          float    ka = sm.key[a], kb = sm.key[b];
          uint32_t ia = sm.idx[a], ib = sm.idx[b];
          const bool gt = (ka > kb) || (ka == kb && ia > ib);
          if (up ? gt : !gt) {
            sm.key[a] = kb; sm.key[b] = ka;
            sm.idx[a] = ib; sm.idx[b] = ia;
          }
        }
      }
      __syncthreads();
    }
  }

  // ---------------- Stage 3: swapping corruption -> stage into sm.key ------
  // Sort keys are dead after Stage 2; reuse their 8 KB as the output buffer.
#pragma unroll
  for (int q = 0; q < EPT; q += 4) {
    const uint32_t col = base + (uint32_t)q;
    uint32_t rs[4];
    philox4(row, col, 2u, 0u, 42u, 0x73776170u, rs);  // "swap"
    float o[4];
#pragma unroll
    for (int t = 0; t < 4; ++t) {
      const uint32_t j = col + (uint32_t)t;
      const bool     sw = (u01(rs[t]) < 0.10f);
      const uint32_t p  = sm.idx[j];           // perm[j]
      o[t] = sw ? sm.val[p] : sm.val[j];       // gather from pre-swap row
    }
    *(float4*)&sm.key[col] = make_float4(o[0], o[1], o[2], o[3]); // ds_store_b128
  }
  __syncthreads();

  // ---------------- Stage 4: row LDS -> Global via Tensor Data Mover -------
#ifdef HAVE_TDM
  if (tid < 32) {  // wave 0 issues the store DMA; other waves retire
    u32x4 g0; i32x8 g1; i32x4 gz = { 0, 0, 0, 0 };
    make_row_desc((unsigned)(FDIM * sizeof(float)),   // sm.key LDS offset
                  (unsigned long long)(void*)(out + rowBase), g0, g1);
#if __has_include(<hip/amd_detail/amd_gfx1250_TDM.h>)
    i32x8 gz8 = { 0, 0, 0, 0, 0, 0, 0, 0 };
    __builtin_amdgcn_tensor_store_from_lds(g0, g1, gz, gz, gz8, 0);
#else
    __builtin_amdgcn_tensor_store_from_lds(g0, g1, gz, gz, 0);
#endif
    __builtin_amdgcn_s_wait_tensorcnt((short)0);  // keep LDS alive until done
  }
#else
  for (uint32_t i = tid; i < FDIM / 4; i += BLOCK) {
    *(float4*)(out + rowBase + 4 * i) = *(const float4*)&sm.key[4 * i];
  }
#endif
}

extern "C" void kernel_launch(void* const* d_in, const int* in_sizes, int n_in,
                              void* d_out, int out_size, void* d_ws, size_t ws_size,
                              hipStream_t stream) {
  (void)n_in; (void)out_size; (void)d_ws; (void)ws_size;
  const float* x   = (const float*)d_in[0];
  float*       out = (float*)d_out;
  const int    B   = in_sizes[0] / FDIM;  // 32768 rows
  corruption_pipeline_7438883357302<<<dim3((unsigned)B), dim3(BLOCK), 0, stream>>>(x, out);
}